// AllegroLayer_10720238371311
// MI455X (gfx1250) — compile-verified
//
#include <hip/hip_runtime.h>
#include <hip/hip_bf16.h>
#include <math.h>

#define N_NODES 16000
#define N_EDGES 256000
#define FDIM 128
#define MUL 64
#define HID 128

constexpr float SQRT3_C      = 1.7320508075688772f;
constexpr float INV_SQRT3_C  = 0.5773502691896258f;
// SILU_C = sqrt(E[silu(z)^2]), z~N(0,1); E = 1/2 - E[z^2 sigma'(z)] ~= 0.355818
constexpr float INV_SILU_C   = 1.6764325f;   // 1 / 0.5965047
constexpr float INV_SQRT128  = 0.08838834764831845f;
constexpr float INV_SQRT256  = 0.0625f;

typedef float v2f __attribute__((ext_vector_type(2)));
typedef float v4f __attribute__((ext_vector_type(4)));
typedef float v8f __attribute__((ext_vector_type(8)));

__device__ __forceinline__ float nsilu(float v) {
  return (v / (1.0f + expf(-v))) * INV_SILU_C;
}

// 16x16 fp32 GEMM tile via V_WMMA_F32_16X16X4_F32, K stepped by 4.
// A: row-major in LDS (lda stride).  Bp: fragment-packed weights in global:
//   Bp[(n*2 + half)*(K/2) + 2*(k/4) + j] = W[k + 2*half + j][n]
// so each lane's B stream is contiguous -> aligned b128 loads, 2 K-steps each.
template <int KDIM>
__device__ __forceinline__ v8f wmma_row_gemm(const float* __restrict__ A, int lda,
                                             const float* __restrict__ Bp,
                                             int nbase, int lane) {
  const int col  = lane & 15;
  const int half = lane >> 4;
  v8f acc = {};
  const float* arow = A + (lane & 15) * lda + 2 * half;
  const float* bs   = Bp + ((size_t)(nbase + col) * 2 + half) * (KDIM / 2);
#pragma unroll
  for (int i = 0; i < KDIM / 4; i += 2) {
    v4f bq = *(const v4f*)(bs + 2 * i);
    v2f a0 = *(const v2f*)(arow + 4 * i);
    v2f a1 = *(const v2f*)(arow + 4 * (i + 1));
    v2f b0 = {bq.x, bq.y};
    v2f b1 = {bq.z, bq.w};
    acc = __builtin_amdgcn_wmma_f32_16x16x4_f32(false, a0, false, b0, (short)0, acc,
                                                false, false);
    acc = __builtin_amdgcn_wmma_f32_16x16x4_f32(false, a1, false, b1, (short)0, acc,
                                                false, false);
  }
  return acc;
}

__global__ void zero_kernel(float* __restrict__ p, int n) {
  int i = blockIdx.x * blockDim.x + threadIdx.x;
  if (i < n) p[i] = 0.0f;
}

__global__ void count_kernel(const int* __restrict__ senders,
                             const float* __restrict__ m,
                             float* __restrict__ node_cnt) {
  int e = blockIdx.x * blockDim.x + threadIdx.x;
  if (e < N_EDGES) atomicAdd(&node_cnt[senders[e]], m[e]);
}

// Repack row-major W[K][N] -> fragment-packed Wp (see wmma_row_gemm).
__global__ void pack_weight_kernel(const float* __restrict__ W,
                                   float* __restrict__ Wp, int K, int N) {
  int idx = blockIdx.x * blockDim.x + threadIdx.x;
  if (idx >= K * N) return;
  int k = idx / N, n = idx - k * N;
  int i = k >> 2, kr = k & 3;
  int half = kr >> 1, j = kr & 1;
  Wp[((size_t)n * 2 + half) * (K / 2) + 2 * i + j] = W[idx];
}

// Kernel 1: w = (x*m) @ W_env / sqrt(128); scatter ws = w0*m and wv = w1*Y1*m
// into node accumulators. One wave = 16 edges.
__global__ void env_scatter_kernel(const float* __restrict__ x,
                                   const float* __restrict__ m,
                                   const float* __restrict__ Wp_env,
                                   const float* __restrict__ vectors,
                                   const int* __restrict__ senders,
                                   float* __restrict__ node_ws,
                                   float* __restrict__ node_wv) {
  extern __shared__ float smem[];
  const int lane = threadIdx.x & 31;
  const int wave = threadIdx.x >> 5;
  float* xm = smem + wave * (16 * FDIM + 64);   // 16x128 staged A tile
  float* y1 = xm + 16 * FDIM;                   // 16x(3 padded to 4)
  const int tile = blockIdx.x * (blockDim.x >> 5) + wave;
  const int base = tile * 16;

  __builtin_prefetch(Wp_env + ((threadIdx.x & 127) << 6), 0, 3);

  for (int idx = lane; idx < 16 * FDIM; idx += 32) {
    int e = idx >> 7;
    xm[idx] = x[(size_t)(base + e) * FDIM + (idx & 127)] * m[base + e];
  }
  if (lane < 16) {
    const float* vv = vectors + (size_t)(base + lane) * 3;
    float vx = vv[0], vy = vv[1], vz = vv[2];
    float s = SQRT3_C / sqrtf(vx * vx + vy * vy + vz * vz);
    y1[lane * 4 + 0] = vx * s;
    y1[lane * 4 + 1] = vy * s;
    y1[lane * 4 + 2] = vz * s;
  }
  const int col = lane & 15, half = lane >> 4;
  for (int nt = 0; nt < 8; ++nt) {
    v8f acc = wmma_row_gemm<FDIM>(xm, FDIM, Wp_env, nt * 16, lane);
    const int n = nt * 16 + col;
#pragma unroll
    for (int r = 0; r < 8; ++r) {
      int mrow = r + 8 * half;
      int e = base + mrow;
      int node = senders[e];
      float wv = acc[r] * INV_SQRT128 * m[e];
      if (n < MUL) {
        atomicAdd(&node_ws[(size_t)node * MUL + n], wv);
      } else {
        int mu = n - MUL;
        float* dst = node_wv + (size_t)node * (3 * MUL) + mu * 3;
        atomicAdd(&dst[0], wv * y1[mrow * 4 + 0]);
        atomicAdd(&dst[1], wv * y1[mrow * 4 + 1]);
        atomicAdd(&dst[2], wv * y1[mrow * 4 + 2]);
      }
    }
  }
}

// Kernel 2: gather wY_s/wY_v, tensor products, 3-layer MLP -> x_out,
// then V_in @ W_out -> V_out. One wave = 16 edges, 24KB LDS union per wave.
__global__ void gather_mlp_vout_kernel(const float* __restrict__ x,
                                       const float* __restrict__ m,
                                       const float* __restrict__ V,
                                       const float* __restrict__ u,
                                       const int* __restrict__ senders,
                                       const float* __restrict__ node_cnt,
                                       const float* __restrict__ node_ws,
                                       const float* __restrict__ node_wv,
                                       const float* __restrict__ Wp1,
                                       const float* __restrict__ Wp2,
                                       const float* __restrict__ Wp3,
                                       const float* __restrict__ WpO,
                                       float* __restrict__ x_out,
                                       float* __restrict__ V_out) {
  extern __shared__ float smem[];
  const int lane = threadIdx.x & 31;
  const int wave = threadIdx.x >> 5;
  float* scr  = smem + wave * 6144;  // 16 x 384 union
  float* xcat = scr;                 // 16 x 256
  float* hbuf = scr + 4096;          // 16 x 128
  const int tile = blockIdx.x * (blockDim.x >> 5) + wave;
  const int base = tile * 16;
  const int col = lane & 15, half = lane >> 4;

  __builtin_prefetch(Wp1 + ((threadIdx.x & 63) << 5), 0, 3);

  // ---- build x_cat = [x*m | s_a | s_b] ----
  for (int idx = lane; idx < 16 * FDIM; idx += 32) {
    int e = idx >> 7;
    xcat[(e << 8) + (idx & 127)] =
        x[(size_t)(base + e) * FDIM + (idx & 127)] * m[base + e];
  }
  for (int idx = lane; idx < 16 * MUL; idx += 32) {
    int e = idx >> 6, mu = idx & 63;
    int ge = base + e;
    int node = senders[ge];
    float inv = 1.0f / (node_cnt[node] + 1e-5f);
    float wys = node_ws[(size_t)node * MUL + mu] * inv;
    float Vs  = V[(size_t)ge * (4 * MUL) + mu];
    float sb = 0.0f;
#pragma unroll
    for (int k = 0; k < 3; ++k) {
      float wyv = node_wv[(size_t)node * (3 * MUL) + mu * 3 + k] * inv;
      float Vv  = V[(size_t)ge * (4 * MUL) + MUL + mu * 3 + k];
      sb += wyv * Vv;
    }
    xcat[(e << 8) + 128 + mu] = wys * Vs;            // s_a
    xcat[(e << 8) + 192 + mu] = sb * INV_SQRT3_C;    // s_b
  }

  // ---- layer 1: (16x256)@(256x128), nsilu ----
  for (int nt = 0; nt < 8; ++nt) {
    v8f acc = wmma_row_gemm<2 * FDIM>(xcat, 256, Wp1, nt * 16, lane);
#pragma unroll
    for (int r = 0; r < 8; ++r)
      hbuf[(r + 8 * half) * HID + nt * 16 + col] = nsilu(acc[r] * INV_SQRT256);
  }
  // ---- layer 2: (16x128)@(128x128), nsilu -> reuse front of scr ----
  for (int nt = 0; nt < 8; ++nt) {
    v8f acc = wmma_row_gemm<HID>(hbuf, HID, Wp2, nt * 16, lane);
#pragma unroll
    for (int r = 0; r < 8; ++r)
      scr[(r + 8 * half) * HID + nt * 16 + col] = nsilu(acc[r] * INV_SQRT128);
  }
  // ---- layer 3: (16x128)@(128x128) * u -> global x_out ----
  for (int nt = 0; nt < 8; ++nt) {
    v8f acc = wmma_row_gemm<HID>(scr, HID, Wp3, nt * 16, lane);
#pragma unroll
    for (int r = 0; r < 8; ++r) {
      int e = base + r + 8 * half;
      x_out[(size_t)e * HID + nt * 16 + col] = u[e] * acc[r] * INV_SQRT128;
    }
  }

  // ---- build V_in = [v_a | v_b] as 16 x 3 x 128 (LDS union reuse) ----
  for (int idx = lane; idx < 16 * MUL; idx += 32) {
    int e = idx >> 6, mu = idx & 63;
    int ge = base + e;
    int node = senders[ge];
    float inv = 1.0f / (node_cnt[node] + 1e-5f);
    float wys = node_ws[(size_t)node * MUL + mu] * inv;
    float Vs  = V[(size_t)ge * (4 * MUL) + mu];
#pragma unroll
    for (int k = 0; k < 3; ++k) {
      float wyv = node_wv[(size_t)node * (3 * MUL) + mu * 3 + k] * inv;
      float Vv  = V[(size_t)ge * (4 * MUL) + MUL + mu * 3 + k];
      scr[e * 384 + k * 128 + mu]      = wys * Vv;   // v_a
      scr[e * 384 + k * 128 + 64 + mu] = wyv * Vs;   // v_b
    }
  }
  // ---- V_out[:, :, k] = V_in[:, :, k] @ W_out / sqrt(128) ----
  for (int k = 0; k < 3; ++k) {
    for (int nt = 0; nt < 4; ++nt) {
      v8f acc = wmma_row_gemm<2 * MUL>(scr + k * 128, 384, WpO, nt * 16, lane);
#pragma unroll
      for (int r = 0; r < 8; ++r) {
        int e = base + r + 8 * half;
        int o = nt * 16 + col;
        V_out[(size_t)e * (3 * MUL) + o * 3 + k] = acc[r] * INV_SQRT128;
      }
    }
  }
}

extern "C" void kernel_launch(void* const* d_in, const int* in_sizes, int n_in,
                              void* d_out, int out_size, void* d_ws, size_t ws_size,
                              hipStream_t stream) {
  const float* vectors = (const float*)d_in[0];
  const float* x       = (const float*)d_in[1];
  const float* V       = (const float*)d_in[2];
  const float* u       = (const float*)d_in[3];
  const float* m       = (const float*)d_in[4];
  const float* W_env   = (const float*)d_in[5];
  const float* W1      = (const float*)d_in[6];
  const float* W2      = (const float*)d_in[7];
  const float* W3      = (const float*)d_in[8];
  const float* W_out   = (const float*)d_in[9];
  const int*   senders = (const int*)d_in[10];

  float* node_cnt = (float*)d_ws;                       // N_NODES
  float* node_ws  = node_cnt + N_NODES;                 // N_NODES*64
  float* node_wv  = node_ws + (size_t)N_NODES * MUL;    // N_NODES*192
  float* Wp_env   = node_wv + (size_t)N_NODES * 3 * MUL;
  float* Wp1      = Wp_env + 128 * 128;
  float* Wp2      = Wp1 + 256 * 128;
  float* Wp3      = Wp2 + 128 * 128;
  float* WpO      = Wp3 + 128 * 128;

  const int nzero = N_NODES * (1 + MUL + 3 * MUL);
  zero_kernel<<<(nzero + 255) / 256, 256, 0, stream>>>(node_cnt, nzero);
  count_kernel<<<(N_EDGES + 255) / 256, 256, 0, stream>>>(senders, m, node_cnt);

  pack_weight_kernel<<<(128 * 128 + 255) / 256, 256, 0, stream>>>(W_env, Wp_env, 128, 128);
  pack_weight_kernel<<<(256 * 128 + 255) / 256, 256, 0, stream>>>(W1, Wp1, 256, 128);
  pack_weight_kernel<<<(128 * 128 + 255) / 256, 256, 0, stream>>>(W2, Wp2, 128, 128);
  pack_weight_kernel<<<(128 * 128 + 255) / 256, 256, 0, stream>>>(W3, Wp3, 128, 128);
  pack_weight_kernel<<<(128 * 64 + 255) / 256, 256, 0, stream>>>(W_out, WpO, 128, 64);

  // 4 waves/block, 16 edges per wave
  const size_t smem1 = 4 * (16 * FDIM + 64) * sizeof(float);  // 33792 B
  env_scatter_kernel<<<N_EDGES / (16 * 4), 128, smem1, stream>>>(
      x, m, Wp_env, vectors, senders, node_ws, node_wv);

  float* x_out = (float*)d_out;
  float* V_out = x_out + (size_t)N_EDGES * HID;
  // 2 waves/block, 24KB LDS per wave
  const size_t smem2 = 2 * 6144 * sizeof(float);  // 49152 B
  gather_mlp_vout_kernel<<<N_EDGES / (16 * 2), 64, smem2, stream>>>(
      x, m, V, u, senders, node_cnt, node_ws, node_wv,
      Wp1, Wp2, Wp3, WpO, x_out, V_out);
}